// BiAttnScorer_11656541241670
// MI455X (gfx1250) — compile-verified
//
#include <hip/hip_runtime.h>
#include <math.h>

// Problem constants (fixed by the reference): B=128, L=512, D=256
#define B_   128
#define L_   512
#define D_   256
#define TEMP 0.1f
#define NEGV -10000.0f
#define EPSV 1e-8f

typedef __attribute__((ext_vector_type(2))) float v2f;
typedef __attribute__((ext_vector_type(8))) float v8f;

// ---------------------------------------------------------------------------
// Kernel 0: init colmax scratch to NEG (atomic-max accumulator target)
// ---------------------------------------------------------------------------
__global__ __launch_bounds__(256) void k_init(float* __restrict__ colmax) {
    colmax[(size_t)blockIdx.x * 256 + threadIdx.x] = NEGV;
}

// ---------------------------------------------------------------------------
// Kernel 1: fused batched GEMM (f32 WMMA 16x16x4) + masked row/col max.
// grid = (4 m-tiles, 128 batches), 256 threads = 8 waves.
// Wave w owns rows [mtile*128 + w*16, +16). Loops all N tiles, K in steps of 4.
//
// WMMA f32 16x16x4 VGPR layouts (ISA 7.12.2):
//   A 16x4 : lane<16 -> (M=lane, K=0/1 in v0/v1); lane>=16 -> (M=lane-16, K=2/3)
//   B 4x16 : lane<16 -> (N=lane, K=0/1);          lane>=16 -> (N=lane-16, K=2/3)
//   D 16x16: vgpr r, lane<16 -> (M=r, N=lane);    lane>=16 -> (M=r+8, N=lane-16)
// => both A and B fragments are a single float2 load per K-step:
//      base + row*D + 2*(lane>>4) + k.
//
// The A strip (16 rows x K=256) is invariant across the N-loop, so it is
// preloaded once into 64 x v2f = 128 VGPRs; the inner loop then issues exactly
// one global_load_b64 (B) per v_wmma. The next B tile is prefetched
// (global_prefetch_b8) before each K-loop to hide L2 latency behind 64 WMMAs.
// ---------------------------------------------------------------------------
__global__ __launch_bounds__(256)
void k_match_max(const float* __restrict__ v1, const int* __restrict__ mask1,
                 const float* __restrict__ v2, const int* __restrict__ mask2,
                 float* __restrict__ rowmax, float* __restrict__ colmax) {
    const int b     = blockIdx.y;
    const int mtile = blockIdx.x;            // 0..3
    const int tid   = threadIdx.x;
    const int wave  = tid >> 5;
    const int lane  = tid & 31;
    const int lo    = lane & 15;
    const int hi    = lane >> 4;

    __shared__ float scol[L_];
    for (int i = tid; i < L_; i += 256) scol[i] = NEGV;
    __syncthreads();

    const int    m0  = mtile * 128 + wave * 16;
    const float* v1b = v1 + (size_t)b * L_ * D_;
    const float* v2b = v2 + (size_t)b * L_ * D_;
    const int*   m1b = mask1 + (size_t)b * L_;
    const int*   m2b = mask2 + (size_t)b * L_;

    // Bitfield of mask1 for this wave's 16 rows: bit j = mask1[m0+j]
    const int      myRow   = m1b[m0 + lo];
    const unsigned rowbits = (unsigned)__ballot(myRow != 0);   // wave32

    // ---- Preload the entire A strip into registers (64 x v2f = 128 VGPRs) --
    const float* aptr = v1b + (size_t)(m0 + lo) * D_ + 2 * hi;
    v2f areg[64];
#pragma unroll
    for (int kk = 0; kk < 64; ++kk)
        areg[kk] = *(const v2f*)(aptr + 4 * kk);

    float rmax[8];
#pragma unroll
    for (int r = 0; r < 8; ++r) rmax[r] = NEGV;

    for (int n0 = 0; n0 < L_; n0 += 16) {
        const float* bptr = v2b + (size_t)(n0 + lo) * D_ + 2 * hi;

        // Prefetch next B tile (one line per lane; lanes hi=0/1 split the row)
        if (n0 + 16 < L_)
            __builtin_prefetch(v2b + (size_t)(n0 + 16 + lo) * D_ + hi * 128, 0, 3);

        v8f acc = {};
#pragma unroll
        for (int kk = 0; kk < 64; ++kk) {
            v2f bv = *(const v2f*)(bptr + 4 * kk);
            // D = A(16x4) x B(4x16) + C, full f32
            acc = __builtin_amdgcn_wmma_f32_16x16x4_f32(
                false, areg[kk], false, bv, (short)0, acc, false, false);
        }

        const int nmask = m2b[n0 + lo];      // this lane's column mask (N = n0+lo)
        float cmax = NEGV;
#pragma unroll
        for (int r = 0; r < 8; ++r) {
            const int rbit = (rowbits >> (r + 8 * hi)) & 1;  // M = m0 + r + 8*hi
            const float val = (rbit && nmask) ? acc[r] : NEGV;
            rmax[r] = fmaxf(rmax[r], val);
            cmax    = fmaxf(cmax, val);
        }
        // merge the two 8-row halves for this column
        cmax = fmaxf(cmax, __shfl_xor(cmax, 16, 32));
        if (lane < 16) atomicMax(&scol[n0 + lane], cmax);     // LDS float max
    }

    // Row max: reduce across the 16 lanes of each half (rows owned exclusively)
#pragma unroll
    for (int r = 0; r < 8; ++r) {
        float v = rmax[r];
        v = fmaxf(v, __shfl_xor(v, 1, 32));
        v = fmaxf(v, __shfl_xor(v, 2, 32));
        v = fmaxf(v, __shfl_xor(v, 4, 32));
        v = fmaxf(v, __shfl_xor(v, 8, 32));
        if (lo == 0) rowmax[(size_t)b * L_ + m0 + r + 8 * hi] = v;
    }

    __syncthreads();
    for (int i = tid; i < L_; i += 256)
        atomicMax(&colmax[(size_t)b * L_ + i], scol[i]);      // global float max
}

// ---------------------------------------------------------------------------
// Kernel 2: masked scores -> softmax -> attention-weighted sums -> cosine sim
// One workgroup (256 threads) per batch. thread t <-> channel d = t (D=256).
// ---------------------------------------------------------------------------
__device__ __forceinline__ float waveMax(float v) {
    v = fmaxf(v, __shfl_xor(v, 16, 32));
    v = fmaxf(v, __shfl_xor(v, 8, 32));
    v = fmaxf(v, __shfl_xor(v, 4, 32));
    v = fmaxf(v, __shfl_xor(v, 2, 32));
    v = fmaxf(v, __shfl_xor(v, 1, 32));
    return v;
}
__device__ __forceinline__ float waveSum(float v) {
    v += __shfl_xor(v, 16, 32);
    v += __shfl_xor(v, 8, 32);
    v += __shfl_xor(v, 4, 32);
    v += __shfl_xor(v, 2, 32);
    v += __shfl_xor(v, 1, 32);
    return v;
}
__device__ __forceinline__ float blockMax(float v, float* red) {
    const int w = threadIdx.x >> 5, l = threadIdx.x & 31;
    v = waveMax(v);
    if (l == 0) red[w] = v;
    __syncthreads();
    float r = red[0];
#pragma unroll
    for (int i = 1; i < 8; ++i) r = fmaxf(r, red[i]);
    __syncthreads();
    return r;
}
__device__ __forceinline__ float blockSum(float v, float* red) {
    const int w = threadIdx.x >> 5, l = threadIdx.x & 31;
    v = waveSum(v);
    if (l == 0) red[w] = v;
    __syncthreads();
    float r = 0.f;
#pragma unroll
    for (int i = 0; i < 8; ++i) r += red[i];
    __syncthreads();
    return r;
}

__global__ __launch_bounds__(256)
void k_attn_cos(const float* __restrict__ v1, const int* __restrict__ mask1,
                const float* __restrict__ v2, const int* __restrict__ mask2,
                const float* __restrict__ rowmax, const float* __restrict__ colmax,
                float* __restrict__ out) {
    const int b   = blockIdx.x;
    const int tid = threadIdx.x;

    __shared__ float a1[L_];   // scores then attention weights (v1 side)
    __shared__ float a2[L_];   // scores then attention weights (v2 side)
    __shared__ float red[8];

    for (int l = tid; l < L_; l += 256) {
        const int m1 = mask1[(size_t)b * L_ + l];
        const int m2 = mask2[(size_t)b * L_ + l];
        a1[l] = m1 ? (-rowmax[(size_t)b * L_ + l] / TEMP) : NEGV;
        a2[l] = m2 ? (-colmax[(size_t)b * L_ + l] / TEMP) : NEGV;
    }
    __syncthreads();

    // stable softmax over L=512 (each thread owns entries tid, tid+256)
    const float mx1 = blockMax(fmaxf(a1[tid], a1[tid + 256]), red);
    float e1a = __expf(a1[tid] - mx1), e1b = __expf(a1[tid + 256] - mx1);
    const float s1 = blockSum(e1a + e1b, red);
    a1[tid] = e1a / s1; a1[tid + 256] = e1b / s1;

    const float mx2 = blockMax(fmaxf(a2[tid], a2[tid + 256]), red);
    float e2a = __expf(a2[tid] - mx2), e2b = __expf(a2[tid + 256] - mx2);
    const float s2 = blockSum(e2a + e2b, red);
    a2[tid] = e2a / s2; a2[tid + 256] = e2b / s2;
    __syncthreads();

    // p1[d] = sum_l attn1[l]*v1[b,l,d]; p2 likewise. d = tid (coalesced).
    // Streaming last-use reads: non-temporal hint (don't promote in caches).
    const float* v1b = v1 + (size_t)b * L_ * D_;
    const float* v2b = v2 + (size_t)b * L_ * D_;
    float p1 = 0.f, p2 = 0.f;
#pragma unroll 4
    for (int l = 0; l < L_; ++l) {
        p1 = fmaf(a1[l], __builtin_nontemporal_load(&v1b[(size_t)l * D_ + tid]), p1);
        p2 = fmaf(a2[l], __builtin_nontemporal_load(&v2b[(size_t)l * D_ + tid]), p2);
    }

    const float dot = blockSum(p1 * p2, red);
    const float nn1 = blockSum(p1 * p1, red);
    const float nn2 = blockSum(p2 * p2, red);
    if (tid == 0) {
        const float n1 = fmaxf(sqrtf(nn1), EPSV);
        const float n2 = fmaxf(sqrtf(nn2), EPSV);
        out[b] = dot / (n1 * n2);
    }
}

// ---------------------------------------------------------------------------
// Launcher
// ---------------------------------------------------------------------------
extern "C" void kernel_launch(void* const* d_in, const int* in_sizes, int n_in,
                              void* d_out, int out_size, void* d_ws, size_t ws_size,
                              hipStream_t stream) {
    const float* v1    = (const float*)d_in[0];
    const int*   mask1 = (const int*)d_in[1];
    const float* v2    = (const float*)d_in[2];
    const int*   mask2 = (const int*)d_in[3];
    float*       out   = (float*)d_out;

    float* rowmax = (float*)d_ws;                 // B*L floats = 256 KB
    float* colmax = rowmax + (size_t)B_ * L_;     // B*L floats = 256 KB

    k_init<<<(B_ * L_) / 256, 256, 0, stream>>>(colmax);
    k_match_max<<<dim3(4, B_), 256, 0, stream>>>(v1, mask1, v2, mask2, rowmax, colmax);
    k_attn_cos<<<B_, 256, 0, stream>>>(v1, mask1, v2, mask2, rowmax, colmax, out);
}